// QCONV1d_orig_35880156791516
// MI455X (gfx1250) — compile-verified
//
#include <hip/hip_runtime.h>

// ---------------------------------------------------------------------------
// QCONV1d on MI455X (gfx1250).
// Closed form of the K=3 VQC:  ev = cos(a1 + w[0,1]) * cos(a2 + w[0,2])
//   vals[b,m] = sum_{t=0..255} cos(x[b, m+(3t+1)/256, (3t+1)%256] + W1)
//                            * cos(x[b, m+(3t+2)/256, (3t+2)%256] + W2)
//   out[b, m, 0..63] = vals[b,m]  (m = 0..252),  out[b, 253, :] = 0
//
// Reduction over the 256 triples via V_WMMA_F32_16X16X4_F32:
//   A[m_local, k] = term,  B = ones(4x16)  ->  D += row sums (full f32).
// D's 16 identical columns feed the 64 broadcast output filters directly.
// Staging: GLOBAL_LOAD_ASYNC_TO_LDS_B128 (ASYNCcnt) when the toolchain
// exposes the builtin, else classic load->ds_store.
// ---------------------------------------------------------------------------

typedef float v2f __attribute__((ext_vector_type(2)));
typedef float v8f __attribute__((ext_vector_type(8)));
typedef int   v4i_ __attribute__((vector_size(16)));   // matches builtin proto

#define W_DIM   256
#define C_DIM   256
#define M_MAX   252      // last valid output row (m_idx = 0..252)
#define OUTW    254      // output width (row 253 is zero padding)
#define LDSROW  260      // padded LDS row stride (floats): 16B aligned, bank-skewed
#define NROWS   66       // rows staged per block (64 m's + 2 window overhang)

#if defined(__has_builtin)
#if __has_builtin(__builtin_amdgcn_global_load_async_to_lds_b128)
#define USE_ASYNC_LDS 1
#endif
#endif

__global__ __launch_bounds__(128)
void qconv_vqc_wmma(const float* __restrict__ x,
                    const float* __restrict__ wts,
                    float* __restrict__ out)
{
    extern __shared__ float lds[];   // NROWS * LDSROW floats (68,640 B, dynamic)

    const int b    = blockIdx.x >> 2;    // 0..31
    const int quad = blockIdx.x & 3;     // 0..3 -> 64 m's each
    const int row0 = quad << 6;          // first staged row of x[b]
    const int tid  = threadIdx.x;

    const float W1 = wts[1];
    const float W2 = wts[2];
    const float* xb = x + (size_t)b * (W_DIM * C_DIM);

    // --- stage rows [row0, row0+66) of x[b] into bank-skewed LDS ------------
    // 66 rows * 64 float4 chunks = 4224 chunks / 128 threads
#ifdef USE_ASYNC_LDS
    for (int i = tid; i < NROWS * 64; i += 128) {
        int r  = i >> 6;                 // 0..65
        int c4 = (i & 63) << 2;          // float column, step 4
        int gr = row0 + r;
        if (gr > W_DIM - 1) gr = W_DIM - 1;       // clamp (duplicate, unused)
        __builtin_amdgcn_global_load_async_to_lds_b128(
            (v4i_*)(xb + ((size_t)gr << 8) + c4),
            (__attribute__((address_space(3))) v4i_*)(&lds[r * LDSROW + c4]),
            0, 0);
    }
#if __has_builtin(__builtin_amdgcn_s_wait_asynccnt)
    __builtin_amdgcn_s_wait_asynccnt(0);
#else
    asm volatile("s_wait_asynccnt 0" ::: "memory");
#endif
    __syncthreads();
#else
    for (int i = tid; i < NROWS * 64; i += 128) {
        int r  = i >> 6;
        int c4 = (i & 63) << 2;
        int gr = row0 + r;
        if (gr > W_DIM - 1) gr = W_DIM - 1;
        const float4 v = *(const float4*)(xb + ((size_t)gr << 8) + c4);
        *(float4*)(&lds[r * LDSROW + c4]) = v;
    }
    __syncthreads();
#endif

    // --- per-wave 16-m tile; WMMA f32 16x16x4 reduction over 256 triples ----
    const int lane   = tid & 31;
    const int wv     = tid >> 5;             // wave in block, 0..3
    const int m_base = row0 + (wv << 4);     // this wave's first m
    const int arow   = lane & 15;            // A-matrix row fed by this lane
    const int kh     = lane >> 4;            // 0: K=0,1   1: K=2,3  (A layout)

    int m = m_base + arow;
    if (m > M_MAX) m = M_MAX;                // clamp invalid rows (not stored)
    const int lrow = m - row0;               // local LDS row of window start

    const v2f bones = {1.0f, 1.0f};          // B = all ones (layout-free)
    v8f acc = {0.f, 0.f, 0.f, 0.f, 0.f, 0.f, 0.f, 0.f};

    for (int i = 0; i < 64; ++i) {
        const int t0 = (i << 2) + (kh << 1);       // this lane's K=even triple
        int j  = 3 * t0 + 1;                       // theta1 element of t0
        float u0 = __builtin_cosf(lds[(lrow + (j >> 8)) * LDSROW + (j & 255)] + W1);
        j += 1;                                    // theta2 element of t0
        float v0 = __builtin_cosf(lds[(lrow + (j >> 8)) * LDSROW + (j & 255)] + W2);
        int j2 = 3 * t0 + 4;                       // theta1 element of t0+1
        float u1 = __builtin_cosf(lds[(lrow + (j2 >> 8)) * LDSROW + (j2 & 255)] + W1);
        j2 += 1;                                   // theta2 element of t0+1
        float v1 = __builtin_cosf(lds[(lrow + (j2 >> 8)) * LDSROW + (j2 & 255)] + W2);

        v2f av = { u0 * v0, u1 * v1 };             // A fragment (2 K-slices)
        // D = A(16x4) x ones(4x16) + C : per-row sum-of-4, accumulated
        acc = __builtin_amdgcn_wmma_f32_16x16x4_f32(
                  false, av, false, bones, (short)0, acc, false, false);
    }

    // --- store: lane holds D[M = v + 8*kh, N = lane&15]; all N identical ----
    // out[b][m][f], f = 4*(lane&15) .. +3, one float4 -> coalesced 256B rows.
    // Guard is wave-uniform except for the last tile (m_base == 240).
    const int colgrp = lane & 15;
    float4* out4 = (float4*)out;
    const size_t obase = (size_t)b * OUTW;
    if (m_base + 15 <= M_MAX) {
        #pragma unroll
        for (int v = 0; v < 8; ++v) {
            const int mo = m_base + v + (kh << 3);
            const float s = acc[v];
            out4[(obase + mo) * 16 + colgrp] = make_float4(s, s, s, s);
        }
    } else {
        #pragma unroll
        for (int v = 0; v < 8; ++v) {
            const int mo = m_base + v + (kh << 3);
            if (mo <= M_MAX) {
                const float s = acc[v];
                out4[(obase + mo) * 16 + colgrp] = make_float4(s, s, s, s);
            }
        }
    }

    // --- zero padding row m = 253 (once per b) ------------------------------
    if (quad == 0 && tid < 16) {
        out4[(obase + 253) * 16 + tid] = make_float4(0.f, 0.f, 0.f, 0.f);
    }
}

// ---------------------------------------------------------------------------
extern "C" void kernel_launch(void* const* d_in, const int* in_sizes, int n_in,
                              void* d_out, int out_size, void* d_ws, size_t ws_size,
                              hipStream_t stream)
{
    const float* x   = (const float*)d_in[0];   // (32, 256, 256) f32
    const float* wts = (const float*)d_in[1];   // (1, 3) f32
    float* out       = (float*)d_out;           // (32, 254, 64) f32

    const dim3 grid(32 * 4);      // (b, quad-of-64-m)
    const dim3 block(128);        // 4 waves; wave = one 16-m WMMA tile
    const size_t shmem = (size_t)NROWS * LDSROW * sizeof(float);  // 68,640 B

    hipLaunchKernelGGL(qconv_vqc_wmma, grid, block, shmem, stream, x, wts, out);
}